// UnsupervisedGAT_7395933683919
// MI455X (gfx1250) — compile-verified
//
#include <hip/hip_runtime.h>
#include <math.h>

#define N_NODES 100000
#define N_EDGES 1600000
#define DIM 128
#define NHEAD 8
#define HDIM 16

typedef float v2f __attribute__((ext_vector_type(2)));
typedef float v8f __attribute__((ext_vector_type(8)));

// ---------------------------------------------------------------------------
// feat = A[nrows,128] @ W[128,128] via V_WMMA_F32_16X16X4_F32.
// One wave per 16-row stripe, 8 accumulator tiles of 16x16 across 128 cols.
// A-fragment (16x4 f32): lanes 0-15 hold K={k0,k0+1}, lanes 16-31 K={k0+2,k0+3}.
// B-fragment (4x16 f32): mirrored row striping across lanes.
// ---------------------------------------------------------------------------
__global__ __launch_bounds__(256) void gat_gemm_wmma(
    const float* __restrict__ A, const float* __restrict__ W,
    float* __restrict__ feat, int nrows)
{
    const int wave = blockIdx.x * (blockDim.x >> 5) + (threadIdx.x >> 5);
    const int lane = threadIdx.x & 31;
    const int r0 = wave * 16;
    if (r0 >= nrows) return;                 // wave-uniform: EXEC stays all-1s

    const int halfSel = lane >> 4;           // 0: lanes 0-15, 1: lanes 16-31
    const int l = lane & 15;
    const int m = r0 + l;

    v8f acc[8];
#pragma unroll
    for (int t = 0; t < 8; ++t) acc[t] = (v8f){0.f,0.f,0.f,0.f,0.f,0.f,0.f,0.f};

    for (int k0 = 0; k0 < DIM; k0 += 4) {
        const int kk = k0 + (halfSel << 1);
        v2f a;
        a.x = A[(size_t)m * DIM + kk];
        a.y = A[(size_t)m * DIM + kk + 1];
#pragma unroll
        for (int t = 0; t < 8; ++t) {
            v2f b;
            b.x = W[kk * DIM + t * 16 + l];
            b.y = W[(kk + 1) * DIM + t * 16 + l];
            acc[t] = __builtin_amdgcn_wmma_f32_16x16x4_f32(
                false, a, false, b, (short)0, acc[t], false, false);
        }
    }

    // D layout: VGPR r -> row M=r (lanes 0-15) / M=8+r (lanes 16-31), col = lane&15
    const int rowOff = halfSel * 8;
#pragma unroll
    for (int t = 0; t < 8; ++t) {
#pragma unroll
        for (int r = 0; r < 8; ++r) {
            feat[(size_t)(r0 + rowOff + r) * DIM + t * 16 + l] = acc[t][r];
        }
    }
}

// el/er: per (node, head) 16-wide dot products with attention vectors
__global__ void gat_elr(const float* __restrict__ feat,
                        const float* __restrict__ al, const float* __restrict__ ar,
                        float* __restrict__ el, float* __restrict__ er)
{
    int tid = blockIdx.x * blockDim.x + threadIdx.x;
    if (tid >= N_NODES * NHEAD) return;
    int n = tid >> 3, h = tid & 7;
    const float* f = feat + (size_t)n * DIM + h * HDIM;
    const float* a = al + h * HDIM;
    const float* b = ar + h * HDIM;
    float sl = 0.f, sr = 0.f;
#pragma unroll
    for (int d = 0; d < HDIM; ++d) { float v = f[d]; sl += v * a[d]; sr += v * b[d]; }
    el[tid] = sl;
    er[tid] = sr;
}

__global__ void gat_zero(float* __restrict__ p, int n)
{
    int tid = blockIdx.x * blockDim.x + threadIdx.x;
    if (tid < n) p[tid] = 0.f;
}

// denom[d,h] += exp(leaky(el[s,h] + er[d,h], 0.2)) -- no max-shift (safe range)
__global__ void gat_edge_denom(const int* __restrict__ src, const int* __restrict__ dst,
                               const float* __restrict__ el, const float* __restrict__ er,
                               float* __restrict__ denom)
{
    int tid = blockIdx.x * blockDim.x + threadIdx.x;
    if (tid >= N_EDGES * NHEAD) return;
    int e = tid >> 3, h = tid & 7;
    int s = src[e], d = dst[e];
    float x = el[s * NHEAD + h] + er[d * NHEAD + h];
    x = (x >= 0.f) ? x : 0.2f * x;
    atomicAdd(&denom[d * NHEAD + h], __expf(x));
}

// rst = h_in + bias (residual + bias pre-initialized, aggregation adds on top)
__global__ void gat_init_rst(const float* __restrict__ hin, const float* __restrict__ bias,
                             float* __restrict__ rst)
{
    int tid = blockIdx.x * blockDim.x + threadIdx.x;
    if (tid >= N_NODES * DIM) return;
    rst[tid] = hin[tid] + bias[tid & (DIM - 1)];
}

// one thread per (edge, 4-float chunk): gather feat[src], scale by softmax
// weight, fp32 atomic scatter-add into rst[dst]
__global__ void gat_aggregate(const int* __restrict__ src, const int* __restrict__ dst,
                              const float* __restrict__ el, const float* __restrict__ er,
                              const float* __restrict__ denom,
                              const float* __restrict__ feat, float* __restrict__ rst)
{
    long long tid = (long long)blockIdx.x * blockDim.x + threadIdx.x;
    if (tid >= (long long)N_EDGES * 32) return;
    int e  = (int)(tid >> 5);
    int q  = (int)(tid & 31);
    int d0 = q << 2;           // starting feature dim of this 4-float chunk
    int h  = d0 >> 4;          // head index
    int s = src[e], d = dst[e];
    float x = el[s * NHEAD + h] + er[d * NHEAD + h];
    x = (x >= 0.f) ? x : 0.2f * x;
    float w = __expf(x) / denom[d * NHEAD + h];
    const float4 f = *(const float4*)(feat + (size_t)s * DIM + d0);
    float* out = rst + (size_t)d * DIM + d0;
    atomicAdd(out + 0, w * f.x);
    atomicAdd(out + 1, w * f.y);
    atomicAdd(out + 2, w * f.z);
    atomicAdd(out + 3, w * f.w);
}

__global__ void gat_leaky(float* __restrict__ x, int n)
{
    int tid = blockIdx.x * blockDim.x + threadIdx.x;
    if (tid >= n) return;
    float v = x[tid];
    x[tid] = (v >= 0.f) ? v : 0.01f * v;
}

static void run_layer(const float* hin, const int* src, const int* dst,
                      const float* W, const float* al, const float* ar, const float* bias,
                      float* feat, float* el, float* er, float* denom, float* out,
                      bool act, hipStream_t stream)
{
    const int waves = N_NODES / 16;                      // 6250 (N divisible by 16)
    gat_gemm_wmma<<<(waves + 7) / 8, 256, 0, stream>>>(hin, W, feat, N_NODES);
    gat_elr<<<(N_NODES * NHEAD + 255) / 256, 256, 0, stream>>>(feat, al, ar, el, er);
    gat_zero<<<(N_NODES * NHEAD + 255) / 256, 256, 0, stream>>>(denom, N_NODES * NHEAD);
    gat_edge_denom<<<(N_EDGES * NHEAD + 255) / 256, 256, 0, stream>>>(src, dst, el, er, denom);
    gat_init_rst<<<(N_NODES * DIM + 255) / 256, 256, 0, stream>>>(hin, bias, out);
    gat_aggregate<<<(int)(((long long)N_EDGES * 32 + 255) / 256), 256, 0, stream>>>(
        src, dst, el, er, denom, feat, out);
    if (act)
        gat_leaky<<<(N_NODES * DIM + 255) / 256, 256, 0, stream>>>(out, N_NODES * DIM);
}

extern "C" void kernel_launch(void* const* d_in, const int* in_sizes, int n_in,
                              void* d_out, int out_size, void* d_ws, size_t ws_size,
                              hipStream_t stream)
{
    const float* n_feat = (const float*)d_in[0];
    const int*   src    = (const int*)d_in[1];
    const int*   dst    = (const int*)d_in[2];
    const float* W0  = (const float*)d_in[3];
    const float* al0 = (const float*)d_in[4];
    const float* ar0 = (const float*)d_in[5];
    const float* b0  = (const float*)d_in[6];
    const float* W1  = (const float*)d_in[7];
    const float* al1 = (const float*)d_in[8];
    const float* ar1 = (const float*)d_in[9];
    const float* b1  = (const float*)d_in[10];

    float* ws    = (float*)d_ws;
    float* feat  = ws;                                   // N*DIM
    float* el    = feat  + (size_t)N_NODES * DIM;        // N*H
    float* er    = el    + (size_t)N_NODES * NHEAD;      // N*H
    float* denom = er    + (size_t)N_NODES * NHEAD;      // N*H
    float* h1    = denom + (size_t)N_NODES * NHEAD;      // N*DIM
    float* out   = (float*)d_out;

    run_layer(n_feat, src, dst, W0, al0, ar0, b0, feat, el, er, denom, h1,  true,  stream);
    run_layer(h1,     src, dst, W1, al1, ar1, b1, feat, el, er, denom, out, false, stream);
}